// STP_GR_Net_6_1202590843144
// MI455X (gfx1250) — compile-verified
//
#include <hip/hip_runtime.h>

typedef __attribute__((ext_vector_type(2))) float v2f;
typedef __attribute__((ext_vector_type(8))) float v8f;

#define N_NODES  50000
#define N_EDGES  1600000
#define N_TILES  3125      // 50000/16 exactly
#define HIST_LEN 16
#define EMB      32
#define ENCD     64
#define DECD     128
#define OUT_LEN  25
#define NB       500
#define HSTR     68        // LDS row stride (floats) for 64-wide h (even -> float2 aligned)
#define ESTR     36        // LDS row stride for 32-wide emb tile
#define DSTR     132       // LDS row stride for 128-wide decoder state

// ---------------- WMMA helpers (fp32 16x16x4) ----------------
__device__ __forceinline__ v8f wmma4(v2f a, v2f b, v8f c) {
  // 8 args: (neg_a, A, neg_b, B, c_mod, C, reuse_a, reuse_b)
  return __builtin_amdgcn_wmma_f32_16x16x4_f32(false, a, false, b, (short)0, c, false, false);
}
__device__ __forceinline__ float lrelu(float v) { return v >= 0.f ? v : 0.1f * v; }
__device__ __forceinline__ float sigf(float v)  { return 1.f / (1.f + __expf(-v)); }
__device__ __forceinline__ v8f splat8(float v) {
  v8f r;
#pragma unroll
  for (int i = 0; i < 8; ++i) r[i] = v;
  return r;
}
__device__ __forceinline__ v8f sig8(v8f v) {
  v8f r;
#pragma unroll
  for (int i = 0; i < 8; ++i) r[i] = sigf(v[i]);
  return r;
}
__device__ __forceinline__ v8f tanh8(v8f v) {
  v8f r;
#pragma unroll
  for (int i = 0; i < 8; ++i) r[i] = tanhf(v[i]);
  return r;
}

// D += A(16xKD from LDS, row stride S) * W^T  (W: torch Linear weight [Nout][KD] row-major)
// Rolled K-loop (unroll 4) => weight-fragment loads stay loop-variant: no LICM hoist, no spills.
template <int KD, int S>
__device__ __forceinline__ v8f gemm_ldsA(const float* __restrict__ A, const float* __restrict__ W,
                                         int Ncol, int m, int kh, v8f acc) {
#pragma unroll 4
  for (int kk = 0; kk < KD / 4; ++kk) {
    const int k0 = kk * 4 + kh * 2;
    v2f a = *(const v2f*)(A + m * S + k0);
    v2f b = *(const v2f*)(W + Ncol * KD + k0);
    acc = wmma4(a, b, acc);
  }
  return acc;
}

// D += X(16x64 from global, row stride 64) * W  (W: [64][64] row-major slab, B[k][n] = W[k*64+n])
__device__ __forceinline__ v8f gemm_g64(const float* __restrict__ X, int rowbase,
                                        const float* __restrict__ W, int Ncol, int m, int kh, v8f acc) {
#pragma unroll 4
  for (int kk = 0; kk < 16; ++kk) {
    const int k0 = kk * 4 + kh * 2;
    v2f a = *(const v2f*)(X + (rowbase + m) * 64 + k0);
    v2f b;
    b.x = W[k0 * 64 + Ncol];
    b.y = W[(k0 + 1) * 64 + Ncol];
    acc = wmma4(a, b, acc);
  }
  return acc;
}

// ---------------- Kernel 1: ip_emb + GRU encode + dyn_emb -> hist_enc [50000,64] ----------------
__global__ __launch_bounds__(256) void k_encode(
    const float* __restrict__ x, const float* __restrict__ W_ip, const float* __restrict__ b_ip,
    const float* __restrict__ Wi, const float* __restrict__ Wh,
    const float* __restrict__ bi, const float* __restrict__ bh,
    const float* __restrict__ W_dyn, const float* __restrict__ b_dyn,
    float* __restrict__ hist) {
  __shared__ __align__(16) float hball[8][16 * HSTR];
  __shared__ __align__(16) float eball[8][16 * ESTR];
  const int wave = threadIdx.x >> 5;
  int tile = blockIdx.x * 8 + wave;
  if (tile >= N_TILES) tile = N_TILES - 1;  // duplicate work, identical writes (keeps barriers uniform)
  float* hb = hball[wave];
  float* eb = eball[wave];
  const int lane = threadIdx.x & 31;
  const int m = lane & 15;
  const int kh = lane >> 4;
  const int nc = m;  // B/C column within a 16-wide tile
  const int node = tile * 16 + m;

  for (int i = lane; i < 16 * HSTR; i += 32) hb[i] = 0.f;
  __syncthreads();

  v8f hn[4];
#pragma unroll
  for (int j = 0; j < 4; ++j) hn[j] = splat8(0.f);

#pragma unroll 1
  for (int t = 0; t < HIST_LEN; ++t) {
    const float x0 = x[(node * HIST_LEN + t) * 2 + 0];
    const float x1 = x[(node * HIST_LEN + t) * 2 + 1];
    // emb = leaky(x @ W_ip^T + b_ip), written to LDS directly in A-fragment-friendly layout.
    // Lane (m, kh) produces emb[m][k0], emb[m][k0+1] for k0 = 4*kk + 2*kh -> full 16x32 tile.
#pragma unroll
    for (int kk = 0; kk < 8; ++kk) {
      const int k0 = kk * 4 + kh * 2;
      v2f e;
      e.x = lrelu(x0 * W_ip[k0 * 2 + 0] + x1 * W_ip[k0 * 2 + 1] + b_ip[k0]);
      e.y = lrelu(x0 * W_ip[(k0 + 1) * 2 + 0] + x1 * W_ip[(k0 + 1) * 2 + 1] + b_ip[k0 + 1]);
      *(v2f*)(eb + m * ESTR + k0) = e;
    }
    __syncthreads();  // emb tile visible
#pragma unroll
    for (int j = 0; j < 4; ++j) {
      const int cR = j * 16 + nc;
      const int cZ = 64 + j * 16 + nc;
      const int cN = 128 + j * 16 + nc;
      // r gate (combined biases), consumed immediately
      v8f acc = gemm_ldsA<EMB, ESTR>(eb, Wi, cR, m, kh, splat8(bi[cR] + bh[cR]));
      acc = gemm_ldsA<ENCD, HSTR>(hb, Wh, cR, m, kh, acc);
      v8f r = sig8(acc);
      // n gate pieces (needs r): gi part and gh part kept separate
      acc = gemm_ldsA<ENCD, HSTR>(hb, Wh, cN, m, kh, splat8(bh[cN]));
      v8f nn = r * acc;  // r * (h@Wh^T + bh)
      acc = gemm_ldsA<EMB, ESTR>(eb, Wi, cN, m, kh, splat8(bi[cN]));
      nn = tanh8(nn + acc);
      // z gate
      acc = gemm_ldsA<EMB, ESTR>(eb, Wi, cZ, m, kh, splat8(bi[cZ] + bh[cZ]));
      acc = gemm_ldsA<ENCD, HSTR>(hb, Wh, cZ, m, kh, acc);
      v8f z = sig8(acc);
      v8f hold;
#pragma unroll
      for (int rr = 0; rr < 8; ++rr) hold[rr] = hb[(rr + 8 * kh) * HSTR + j * 16 + nc];
      hn[j] = (splat8(1.f) - z) * nn + z * hold;
    }
    __syncthreads();  // all reads of old h done
#pragma unroll
    for (int j = 0; j < 4; ++j) {
#pragma unroll
      for (int rr = 0; rr < 8; ++rr) hb[(rr + 8 * kh) * HSTR + j * 16 + nc] = hn[j][rr];
    }
    __syncthreads();  // new h visible
  }

  // hist_enc = leaky( leaky(h) @ W_dyn^T + b_dyn )
#pragma unroll
  for (int j = 0; j < 4; ++j) {
#pragma unroll
    for (int rr = 0; rr < 8; ++rr) hb[(rr + 8 * kh) * HSTR + j * 16 + nc] = lrelu(hn[j][rr]);
  }
  __syncthreads();
#pragma unroll
  for (int j = 0; j < 4; ++j) {
    v8f acc = gemm_ldsA<ENCD, HSTR>(hb, W_dyn, j * 16 + nc, m, kh, splat8(b_dyn[j * 16 + nc]));
#pragma unroll
    for (int rr = 0; rr < 8; ++rr)
      hist[(tile * 16 + rr + 8 * kh) * ENCD + j * 16 + nc] = lrelu(acc[rr]);
  }
}

// ---------------- GCN feature GEMMs ----------------
__global__ __launch_bounds__(256) void k_xw1(const float* __restrict__ X, const float* __restrict__ W,
                                             float* __restrict__ out) {
  int tile = blockIdx.x * 8 + (threadIdx.x >> 5);
  if (tile >= N_TILES) tile = N_TILES - 1;
  const int lane = threadIdx.x & 31, m = lane & 15, kh = lane >> 4, nc = m;
#pragma unroll 1
  for (int j = 0; j < 4; ++j) {
    v8f acc = gemm_g64(X, tile * 16, W, j * 16 + nc, m, kh, splat8(0.f));
#pragma unroll
    for (int rr = 0; rr < 8; ++rr)
      out[(tile * 16 + rr + 8 * kh) * ENCD + j * 16 + nc] = acc[rr];
  }
}

__global__ __launch_bounds__(256) void k_xw2(const float* __restrict__ Xa, const float* __restrict__ Xb,
                                             const float* __restrict__ W, float* __restrict__ out) {
  int tile = blockIdx.x * 8 + (threadIdx.x >> 5);
  if (tile >= N_TILES) tile = N_TILES - 1;
  const int lane = threadIdx.x & 31, m = lane & 15, kh = lane >> 4, nc = m;
#pragma unroll 1
  for (int j = 0; j < 4; ++j) {
    v8f acc = gemm_g64(Xa, tile * 16, W, j * 16 + nc, m, kh, splat8(0.f));   // K rows 0..63
    acc = gemm_g64(Xb, tile * 16, W + 64 * ENCD, j * 16 + nc, m, kh, acc);   // K rows 64..127
#pragma unroll
    for (int rr = 0; rr < 8; ++rr)
      out[(tile * 16 + rr + 8 * kh) * ENCD + j * 16 + nc] = acc[rr];
  }
}

// ---------------- Graph aggregation utility kernels ----------------
__global__ void k_zero(float* __restrict__ p, long long n) {
  long long t = (long long)blockIdx.x * blockDim.x + threadIdx.x;
  if (t < n) p[t] = 0.f;
}
__global__ void k_fill(float* __restrict__ p, float v, int n) {
  int t = blockIdx.x * blockDim.x + threadIdx.x;
  if (t < n) p[t] = v;
}
__global__ void k_degree(const int* __restrict__ dst, float* __restrict__ deg, int n) {
  int e = blockIdx.x * blockDim.x + threadIdx.x;
  if (e < n) atomicAdd(&deg[dst[e]], 1.0f);
}
__global__ void k_dinv(const float* __restrict__ deg, float* __restrict__ dinv, int n) {
  int t = blockIdx.x * blockDim.x + threadIdx.x;
  if (t < n) dinv[t] = rsqrtf(deg[t]);
}
__global__ void k_scatter(const float* __restrict__ xw, const int* __restrict__ src,
                          const int* __restrict__ dst, const float* __restrict__ dinv,
                          float* __restrict__ agg, int nEdges) {
  long long t = (long long)blockIdx.x * blockDim.x + threadIdx.x;
  int e = (int)(t >> 5);
  if (e >= nEdges) return;
  int c = (int)(t & 31) * 2;  // each lane handles 2 of 64 columns
  int s = src[e], d = dst[e];
  float coef = dinv[s] * dinv[d];
  atomicAdd(&agg[d * 64 + c], xw[s * 64 + c] * coef);
  atomicAdd(&agg[d * 64 + c + 1], xw[s * 64 + c + 1] * coef);
}
__global__ void k_combine(const float* __restrict__ agg, const float* __restrict__ xw,
                          const float* __restrict__ dinv, const float* __restrict__ b,
                          float* __restrict__ g, long long n) {
  long long t = (long long)blockIdx.x * blockDim.x + threadIdx.x;
  if (t >= n) return;
  int i = (int)(t >> 6), c = (int)(t & 63);
  float di = dinv[i];
  g[t] = agg[t] + xw[t] * di * di + b[c];
}

// ---------------- Decoder: target gather + nbr_emb + 2-layer LSTM + out proj ----------------
// 256 threads/block; wave w owns column slab j = w (16 of 128 cols). Cell state per slab in regs.
__global__ __launch_bounds__(256) void k_decode(
    const float* __restrict__ hist, const float* __restrict__ g1, const float* __restrict__ g2,
    const int* __restrict__ tgt,
    const float* __restrict__ W_nbr, const float* __restrict__ b_nbr,
    const float* __restrict__ Wi0, const float* __restrict__ Wh0,
    const float* __restrict__ bi0, const float* __restrict__ bh0,
    const float* __restrict__ Wi1, const float* __restrict__ Wh1,
    const float* __restrict__ bi1, const float* __restrict__ bh1,
    const float* __restrict__ W_op, const float* __restrict__ b_op,
    float* __restrict__ out) {
  __shared__ __align__(16) float enc[16 * DSTR];
  __shared__ __align__(16) float h0[16 * DSTR];
  __shared__ __align__(16) float h1[16 * DSTR];
  __shared__ __align__(16) float htmp[16 * DSTR];
  const int tile = blockIdx.x;
  const int wave = threadIdx.x >> 5;   // slab index j = wave (0..7)
  const int lane = threadIdx.x & 31, m = lane & 15, kh = lane >> 4, nc = m;
  const int j = wave;

  // stage tgt_feat = [g1|g2][target] into htmp; enc low half = hist[target]; zero h-states
  for (int idx = threadIdx.x; idx < 16 * DECD; idx += 256) {
    const int row = idx >> 7, col = idx & 127;
    const int b = tile * 16 + row;
    const int node = (b < NB) ? tgt[b] : 0;
    htmp[row * DSTR + col] = (col < 64) ? g1[node * 64 + col] : g2[node * 64 + col - 64];
    enc[row * DSTR + col] = (col < 64) ? hist[node * 64 + col] : 0.f;
    h0[row * DSTR + col] = 0.f;
    h1[row * DSTR + col] = 0.f;
  }
  __syncthreads();

  // gat_enc = leaky(tgt_feat @ W_nbr^T + b_nbr) -> enc columns 64..127 (waves 0..3, one tile each)
  if (wave < 4) {
    v8f acc = gemm_ldsA<DECD, DSTR>(htmp, W_nbr, j * 16 + nc, m, kh, splat8(b_nbr[j * 16 + nc]));
#pragma unroll
    for (int rr = 0; rr < 8; ++rr)
      enc[(rr + 8 * kh) * DSTR + 64 + j * 16 + nc] = lrelu(acc[rr]);
  }
  __syncthreads();

  v8f c0w = splat8(0.f);  // cell state, layer 0, slab j
  v8f c1w = splat8(0.f);  // cell state, layer 1, slab j
  const int cI = 0 * DECD + j * 16 + nc;
  const int cF = 1 * DECD + j * 16 + nc;
  const int cG = 2 * DECD + j * 16 + nc;
  const int cO = 3 * DECD + j * 16 + nc;

#pragma unroll 1
  for (int step = 0; step < OUT_LEN; ++step) {
    // ---- layer 0: input enc, state (h0, c0w) ----
    {
      v8f acc = gemm_ldsA<DECD, DSTR>(enc, Wi0, cI, m, kh, splat8(bi0[cI] + bh0[cI]));
      acc = gemm_ldsA<DECD, DSTR>(h0, Wh0, cI, m, kh, acc);
      v8f si = sig8(acc);
      acc = gemm_ldsA<DECD, DSTR>(enc, Wi0, cG, m, kh, splat8(bi0[cG] + bh0[cG]));
      acc = gemm_ldsA<DECD, DSTR>(h0, Wh0, cG, m, kh, acc);
      v8f part = si * tanh8(acc);  // sig(i)*tanh(g)
      acc = gemm_ldsA<DECD, DSTR>(enc, Wi0, cF, m, kh, splat8(bi0[cF] + bh0[cF]));
      acc = gemm_ldsA<DECD, DSTR>(h0, Wh0, cF, m, kh, acc);
      c0w = sig8(acc) * c0w + part;
      acc = gemm_ldsA<DECD, DSTR>(enc, Wi0, cO, m, kh, splat8(bi0[cO] + bh0[cO]));
      acc = gemm_ldsA<DECD, DSTR>(h0, Wh0, cO, m, kh, acc);
      v8f hh = sig8(acc) * tanh8(c0w);
#pragma unroll
      for (int rr = 0; rr < 8; ++rr) htmp[(rr + 8 * kh) * DSTR + j * 16 + nc] = hh[rr];
    }
    __syncthreads();
    for (int idx = threadIdx.x; idx < 16 * DECD; idx += 256) {
      int row = idx >> 7, col = idx & 127;
      h0[row * DSTR + col] = htmp[row * DSTR + col];
    }
    __syncthreads();

    // ---- layer 1: input h0, state (h1, c1w) ----
    {
      v8f acc = gemm_ldsA<DECD, DSTR>(h0, Wi1, cI, m, kh, splat8(bi1[cI] + bh1[cI]));
      acc = gemm_ldsA<DECD, DSTR>(h1, Wh1, cI, m, kh, acc);
      v8f si = sig8(acc);
      acc = gemm_ldsA<DECD, DSTR>(h0, Wi1, cG, m, kh, splat8(bi1[cG] + bh1[cG]));
      acc = gemm_ldsA<DECD, DSTR>(h1, Wh1, cG, m, kh, acc);
      v8f part = si * tanh8(acc);
      acc = gemm_ldsA<DECD, DSTR>(h0, Wi1, cF, m, kh, splat8(bi1[cF] + bh1[cF]));
      acc = gemm_ldsA<DECD, DSTR>(h1, Wh1, cF, m, kh, acc);
      c1w = sig8(acc) * c1w + part;
      acc = gemm_ldsA<DECD, DSTR>(h0, Wi1, cO, m, kh, splat8(bi1[cO] + bh1[cO]));
      acc = gemm_ldsA<DECD, DSTR>(h1, Wh1, cO, m, kh, acc);
      v8f hh = sig8(acc) * tanh8(c1w);
#pragma unroll
      for (int rr = 0; rr < 8; ++rr) htmp[(rr + 8 * kh) * DSTR + j * 16 + nc] = hh[rr];
    }
    __syncthreads();
    for (int idx = threadIdx.x; idx < 16 * DECD; idx += 256) {
      int row = idx >> 7, col = idx & 127;
      h1[row * DSTR + col] = htmp[row * DSTR + col];
    }
    __syncthreads();

    // ---- output projection: out[b, step, :] = h1 @ W_op^T + b_op (wave 0 only) ----
    if (wave == 0) {
      float acc = b_op[kh];  // kh = output column (0/1)
      for (int k = 0; k < DECD; ++k) acc += h1[m * DSTR + k] * W_op[kh * DECD + k];
      const int b = tile * 16 + m;
      if (b < NB) out[(b * OUT_LEN + step) * 2 + kh] = acc;
    }
    __syncthreads();
  }
}

// ---------------- Host launch ----------------
extern "C" void kernel_launch(void* const* d_in, const int* in_sizes, int n_in,
                              void* d_out, int out_size, void* d_ws, size_t ws_size,
                              hipStream_t stream) {
  const float* x     = (const float*)d_in[0];
  const float* W_ip  = (const float*)d_in[1];
  const float* b_ip  = (const float*)d_in[2];
  const float* gWi   = (const float*)d_in[3];
  const float* gWh   = (const float*)d_in[4];
  const float* gbi   = (const float*)d_in[5];
  const float* gbh   = (const float*)d_in[6];
  const float* W_dyn = (const float*)d_in[7];
  const float* b_dyn = (const float*)d_in[8];
  const float* W_g1  = (const float*)d_in[9];
  const float* b_g1  = (const float*)d_in[10];
  const float* W_g2  = (const float*)d_in[11];
  const float* b_g2  = (const float*)d_in[12];
  const float* W_nbr = (const float*)d_in[13];
  const float* b_nbr = (const float*)d_in[14];
  const float* Wi0   = (const float*)d_in[15];
  const float* Wh0   = (const float*)d_in[16];
  const float* bi0   = (const float*)d_in[17];
  const float* bh0   = (const float*)d_in[18];
  const float* Wi1   = (const float*)d_in[19];
  const float* Wh1   = (const float*)d_in[20];
  const float* bi1   = (const float*)d_in[21];
  const float* bh1   = (const float*)d_in[22];
  const float* W_op  = (const float*)d_in[23];
  const float* b_op  = (const float*)d_in[24];
  const int* edges   = (const int*)d_in[25];     // [2, N_EDGES]: src then dst
  const int* tgt     = (const int*)d_in[26];
  float* out = (float*)d_out;

  const long long NH = (long long)N_NODES * ENCD;  // 3.2M floats
  float* ws   = (float*)d_ws;
  float* hist = ws + 0 * NH;
  float* xw   = ws + 1 * NH;   // reused for both GCN layers
  float* agg1 = ws + 2 * NH;
  float* agg2 = ws + 3 * NH;   // adjacent to agg1 -> single zero pass
  float* g1   = ws + 4 * NH;
  float* g2   = ws + 5 * NH;
  float* deg  = ws + 6 * NH;
  float* dinv = deg + 50048;

  const int* esrc = edges;
  const int* edst = edges + N_EDGES;

  const int encBlocks = (N_TILES + 7) / 8;  // 391
  const long long scatThreads = (long long)N_EDGES * 32;

  // init (deterministic per call)
  k_zero<<<(int)((2 * NH + 255) / 256), 256, 0, stream>>>(agg1, 2 * NH);
  k_fill<<<(N_NODES + 255) / 256, 256, 0, stream>>>(deg, 1.0f, N_NODES);

  // encoder
  k_encode<<<encBlocks, 256, 0, stream>>>(x, W_ip, b_ip, gWi, gWh, gbi, gbh, W_dyn, b_dyn, hist);

  // degrees / normalization
  k_degree<<<(N_EDGES + 255) / 256, 256, 0, stream>>>(edst, deg, N_EDGES);
  k_dinv<<<(N_NODES + 255) / 256, 256, 0, stream>>>(deg, dinv, N_NODES);

  // GCN layer 1
  k_xw1<<<encBlocks, 256, 0, stream>>>(hist, W_g1, xw);
  k_scatter<<<(int)((scatThreads + 255) / 256), 256, 0, stream>>>(xw, esrc, edst, dinv, agg1, N_EDGES);
  k_combine<<<(int)((NH + 255) / 256), 256, 0, stream>>>(agg1, xw, dinv, b_g1, g1, NH);

  // GCN layer 2 (input = concat[hist, g1])
  k_xw2<<<encBlocks, 256, 0, stream>>>(hist, g1, W_g2, xw);
  k_scatter<<<(int)((scatThreads + 255) / 256), 256, 0, stream>>>(xw, esrc, edst, dinv, agg2, N_EDGES);
  k_combine<<<(int)((NH + 255) / 256), 256, 0, stream>>>(agg2, xw, dinv, b_g2, g2, NH);

  // decoder (32 tiles of 16 targets, 8 waves each = one per column slab)
  k_decode<<<32, 256, 0, stream>>>(hist, g1, g2, tgt, W_nbr, b_nbr,
                                   Wi0, Wh0, bi0, bh0, Wi1, Wh1, bi1, bh1,
                                   W_op, b_op, out);
}